// Model_1236950581277
// MI455X (gfx1250) — compile-verified
//
#include <hip/hip_runtime.h>
#include <hip/hip_bf16.h>
#include <math.h>

// ---------------------------------------------------------------------------
// Mamba classifier for MI455X (gfx1250, wave32).
// ~150 GFLOP fp32 vs ~1 GB HBM traffic -> bandwidth bound. All big GEMMs use
// V_WMMA_F32_16X16X4_F32 with Tensor-Data-Mover (TDM) double-buffered LDS
// staging (tensor_load_to_lds + s_wait_tensorcnt), elementwise stages fused,
// selective scan parallelized over B*D independent 4-state recurrences.
// ---------------------------------------------------------------------------

typedef __attribute__((ext_vector_type(2))) float v2f;
typedef __attribute__((ext_vector_type(8))) float v8f;
typedef unsigned int u32x4 __attribute__((ext_vector_type(4)));
typedef int i32x4 __attribute__((ext_vector_type(4)));
typedef int i32x8 __attribute__((ext_vector_type(8)));

#define KC    16   // K-chunk per TDM stage (divides 256, 176, 128, 64)
#define PITCH 17   // LDS row pitch in floats (TDM pad: +1 DWORD per 16 DWORDs)

__device__ __forceinline__ float silu_f(float v) { return v / (1.0f + expf(-v)); }

// Low 32 bits of a generic pointer into LDS == LDS byte offset (ISA 10.2:
// LDS aperture address is {SHARED_BASE, offset[31:0]}).
__device__ __forceinline__ unsigned lds_off(const void* p) {
    return (unsigned)(unsigned long long)(size_t)p;
}

// ---------------------------------------------------------------------------
// TDM: DMA a rows x KC f32 tile (row-major, source pitch srcPitch floats)
// into LDS at ldsOff with destination pitch 17 floats (pad_interval=16 DWORDs,
// pad_amount=1 DWORD).  D# layout per CDNA5 ISA 8.3/8.4.
// ---------------------------------------------------------------------------
__device__ __forceinline__ void tdm_load_tile(const float* gsrc, unsigned ldsOff,
                                              int rows, int srcPitch) {
    const unsigned long long g = (unsigned long long)(size_t)gsrc;
    u32x4 g0 = { 1u,                                   // count=1, user descriptor
                 ldsOff,                               // lds_addr (bytes)
                 (unsigned)(g & 0xffffffffull),        // global_addr[31:0]
                 (unsigned)((g >> 32) & 0x1ffffffull)  // global_addr[56:32]
                     | (2u << 30) };                   // type=2 ("image")
    const unsigned d0 = (unsigned)KC;     // tensor_dim0 (elements)
    const unsigned d1 = (unsigned)rows;   // tensor_dim1
    const unsigned long long s0 = (unsigned long long)(unsigned)srcPitch; // dim0 stride
    i32x8 g1;
    g1[0] = (int)((2u << 16)     // data_size = 4 bytes
                | (1u << 20)     // pad_enable
                | (3u << 22)     // pad_interval: 2^(3+1) = 16 DWORDs
                | (0u << 25));   // pad_amount: 1 DWORD
    g1[1] = (int)((d0 & 0xffffu) << 16);                            // tensor_dim0[15:0]
    g1[2] = (int)(((d0 >> 16) & 0xffffu) | ((d1 & 0xffffu) << 16)); // dim0 hi | dim1 lo
    g1[3] = (int)(((d1 >> 16) & 0xffffu) | ((unsigned)KC << 16));   // dim1 hi | tile_dim0
    g1[4] = (int)(d1 & 0xffffu);                                    // tile_dim1 (tile_dim2=0)
    g1[5] = (int)(s0 & 0xffffffffull);                              // dim0_stride[31:0]
    g1[6] = (int)((s0 >> 32) & 0xffffull);                          // dim0_stride[47:32]
    g1[7] = 0;                                                      // dim1_stride unused (2D)
    i32x4 z4 = { 0, 0, 0, 0 };
#if defined(__clang_major__) && (__clang_major__ >= 23)
    i32x8 z8 = { 0, 0, 0, 0, 0, 0, 0, 0 };
    __builtin_amdgcn_tensor_load_to_lds(g0, g1, z4, z4, z8, 0);
#else
    __builtin_amdgcn_tensor_load_to_lds(g0, g1, z4, z4, 0);
#endif
}

// ---------------------------------------------------------------------------
// WMMA GEMM with TDM-staged LDS tiles: C[M,N] = A[M,K] @ W[N,K]^T (+bias).
// Block = 8 waves stacked along M (block tile 256 x 16*NT); wave -> 32 x 16*NT.
// Fragment layouts per ISA 7.12.2 (f32 16x16x4):
//   A: lane m = lane&15, K pair {0,1}/{2,3} by lane[4];  B: same with n = lane&15
//   C/D: v8f, row = r + 8*(lane>=16), col = lane&15
// Requires M%256==0, N%(16*NT)==0, K%16==0 (true for every GEMM here).
// ---------------------------------------------------------------------------
template <int NT>
__global__ void gemm_wmma_tdm(const float* __restrict__ A, const float* __restrict__ Wt,
                              const float* __restrict__ bias, float* __restrict__ C,
                              int M, int N, int K, int tilesN) {
    __shared__ float As[2][256 * PITCH];
    __shared__ float Bs[2][32 * PITCH];
    const int bm   = (int)blockIdx.x / tilesN;
    const int bn   = (int)blockIdx.x % tilesN;
    const int w    = (int)(threadIdx.x >> 5);
    const int lane = (int)(threadIdx.x & 31);
    const int row0 = bm * 256;
    const int col0 = bn * (16 * NT);
    const int mr   = lane & 15;
    const int koff = (lane >> 4) << 1;  // 0 or 2
    const int nchunks = K / KC;

    const unsigned aOff = lds_off(&As[0][0]);
    const unsigned bOff = lds_off(&Bs[0][0]);
    const unsigned aBufBytes = 256u * PITCH * 4u;
    const unsigned bBufBytes = 32u * PITCH * 4u;

    if (w == 0) {  // one wave per workgroup drives the DMA engine
        tdm_load_tile(A + (size_t)row0 * K, aOff, 256, K);
        tdm_load_tile(Wt + (size_t)col0 * K, bOff, 16 * NT, K);
    }

    v8f acc00 = {}; v8f acc10 = {}; v8f acc01 = {}; v8f acc11 = {};

    for (int c = 0; c < nchunks; ++c) {
        const int buf = c & 1;
        if (w == 0) __builtin_amdgcn_s_wait_tensorcnt(0);  // chunk c landed in LDS
        __syncthreads();                                   // publish to all waves
        if (w == 0 && (c + 1) < nchunks) {                 // prefetch chunk c+1
            const int nb = (c + 1) & 1;
            tdm_load_tile(A + (size_t)row0 * K + (size_t)(c + 1) * KC,
                          aOff + (unsigned)nb * aBufBytes, 256, K);
            tdm_load_tile(Wt + (size_t)col0 * K + (size_t)(c + 1) * KC,
                          bOff + (unsigned)nb * bBufBytes, 16 * NT, K);
        }
        const float* Ab  = As[buf];
        const float* Bb  = Bs[buf];
        const float* a0p = Ab + (w * 32 + mr) * PITCH + koff;
        const float* a1p = Ab + (w * 32 + 16 + mr) * PITCH + koff;
        const float* b0p = Bb + mr * PITCH + koff;
        const float* b1p = Bb + (16 + mr) * PITCH + koff;
#pragma unroll
        for (int kk = 0; kk < KC; kk += 4) {
            v2f a0 = { a0p[kk], a0p[kk + 1] };
            v2f a1 = { a1p[kk], a1p[kk + 1] };
            v2f b0 = { b0p[kk], b0p[kk + 1] };
            acc00 = __builtin_amdgcn_wmma_f32_16x16x4_f32(false, a0, false, b0, (short)0, acc00, false, false);
            acc10 = __builtin_amdgcn_wmma_f32_16x16x4_f32(false, a1, false, b0, (short)0, acc10, false, false);
            if (NT == 2) {
                v2f b1 = { b1p[kk], b1p[kk + 1] };
                acc01 = __builtin_amdgcn_wmma_f32_16x16x4_f32(false, a0, false, b1, (short)0, acc01, false, false);
                acc11 = __builtin_amdgcn_wmma_f32_16x16x4_f32(false, a1, false, b1, (short)0, acc11, false, false);
            }
        }
        // no trailing barrier needed: chunk c's buffer is only overwritten by the
        // prefetch issued after the *next* iteration's barrier.
    }

    const int cc   = lane & 15;
    const int rsel = (lane >> 4) << 3;  // 0 or 8
    const int wrow = row0 + w * 32;
    const float bv0 = bias ? bias[col0 + cc] : 0.0f;
    const float bv1 = (NT == 2 && bias) ? bias[col0 + 16 + cc] : 0.0f;
#pragma unroll
    for (int r = 0; r < 8; ++r) {
        const int rr = rsel + r;
        C[(size_t)(wrow + rr)      * N + col0 + cc] = acc00[r] + bv0;
        C[(size_t)(wrow + 16 + rr) * N + col0 + cc] = acc10[r] + bv0;
        if (NT == 2) {
            C[(size_t)(wrow + rr)      * N + col0 + 16 + cc] = acc01[r] + bv1;
            C[(size_t)(wrow + 16 + rr) * N + col0 + 16 + cc] = acc11[r] + bv1;
        }
    }
}

// ---- lin1: h = leaky(bn(x @ W1^T + b1)) -----------------------------------
__global__ void lin1_kernel(const float* __restrict__ x, const float* __restrict__ W,
                            const float* __restrict__ bias, const float* __restrict__ bng,
                            const float* __restrict__ bnb, float* __restrict__ h,
                            int Bb, int Ll, int Cin, int Dm) {
    const long long idx = (long long)blockIdx.x * blockDim.x + threadIdx.x;
    const long long total = (long long)Bb * Ll * Dm;
    if (idx >= total) return;
    const int o = (int)(idx % Dm);
    const int l = (int)((idx / Dm) % Ll);
    const long long row = idx / Dm;
    const float* xr = x + row * Cin;
    const float* wr = W + (size_t)o * Cin;
    float acc = bias[o];
    for (int c = 0; c < Cin; ++c) acc += xr[c] * wr[c];
    acc = acc * bng[l] + bnb[l];
    h[idx] = (acc >= 0.0f) ? acc : 0.01f * acc;
}

// ---- LayerNorm over last dim (one wave per row) ---------------------------
__global__ void layernorm_rows(const float* __restrict__ x, const float* __restrict__ g,
                               const float* __restrict__ b, float* __restrict__ out,
                               int rows, int D) {
    const int row  = (int)(blockIdx.x * (blockDim.x >> 5) + (threadIdx.x >> 5));
    const int lane = (int)(threadIdx.x & 31);
    if (row >= rows) return;
    const float* xr = x + (size_t)row * D;
    float s = 0.0f, s2 = 0.0f;
    for (int i = lane; i < D; i += 32) { float v = xr[i]; s += v; s2 += v * v; }
    for (int off = 16; off > 0; off >>= 1) { s += __shfl_xor(s, off); s2 += __shfl_xor(s2, off); }
    const float mean = s / (float)D;
    const float var  = s2 / (float)D - mean * mean;
    const float inv  = rsqrtf(var + 1e-5f);
    float* orow = out + (size_t)row * D;
    for (int i = lane; i < D; i += 32) orow[i] = (xr[i] - mean) * inv * g[i] + b[i];
}

// ---- causal depthwise conv (k=2) + bias + SiLU; reads xi half of xz -------
__global__ void conv_silu(const float* __restrict__ xz, const float* __restrict__ cw,
                          const float* __restrict__ cb, float* __restrict__ xc,
                          int Bb, int Ll, int Dd) {
    const long long idx = (long long)blockIdx.x * blockDim.x + threadIdx.x;
    const long long total = (long long)Bb * Ll * Dd;
    if (idx >= total) return;
    const int d = (int)(idx % Dd);
    const long long r = idx / Dd;
    const int l = (int)(r % Ll);
    const float cur  = xz[r * (2 * (long long)Dd) + d];
    const float prev = (l == 0) ? 0.0f : xz[(r - 1) * (2 * (long long)Dd) + d];
    const float v = prev * cw[d * 2 + 0] + cur * cw[d * 2 + 1] + cb[d];
    xc[idx] = silu_f(v);
}

// ---- x_proj: dbl[r, j] = xc[r,:] . x_proj[j,:]  (NP = dt_rank + 8) --------
__global__ void xproj_kernel(const float* __restrict__ xc, const float* __restrict__ W,
                             float* __restrict__ dbl, long long rows, int D, int NP) {
    const long long idx = (long long)blockIdx.x * blockDim.x + threadIdx.x;
    if (idx >= rows * NP) return;
    const int j = (int)(idx % NP);
    const long long r = idx / NP;
    const float* xr = xc + r * D;
    const float* wr = W + (size_t)j * D;
    float acc = 0.0f;
    for (int k = 0; k < D; ++k) acc += xr[k] * wr[k];
    dbl[idx] = acc;
}

// ---- delta = softplus(dt @ dt_w^T + dt_b) ---------------------------------
__global__ void delta_kernel(const float* __restrict__ dbl, const float* __restrict__ dtw,
                             const float* __restrict__ dtb, float* __restrict__ delta,
                             long long rows, int NP, int dtRank, int D) {
    const long long idx = (long long)blockIdx.x * blockDim.x + threadIdx.x;
    if (idx >= rows * D) return;
    const int d = (int)(idx % D);
    const long long r = idx / D;
    const float* dr = dbl + r * NP;
    const float* wr = dtw + (size_t)d * dtRank;
    float acc = dtb[d];
    for (int j = 0; j < dtRank; ++j) acc += dr[j] * wr[j];
    delta[idx] = (acc > 20.0f) ? acc : log1pf(expf(acc));
}

// ---- selective scan, fused with D-skip and SiLU(z) gate -------------------
__global__ void mamba_scan(const float* __restrict__ delta, const float* __restrict__ xc,
                           const float* __restrict__ dbl, const float* __restrict__ xz,
                           const float* __restrict__ A_log, const float* __restrict__ Dp,
                           float* __restrict__ y, int Bb, int Ll, int Dd, int NP, int dtRank) {
    const int t = (int)(blockIdx.x * blockDim.x + threadIdx.x);
    if (t >= Bb * Dd) return;
    const int b = t / Dd;
    const int d = t - b * Dd;
    const float A0 = -expf(A_log[d * 4 + 0]);
    const float A1 = -expf(A_log[d * 4 + 1]);
    const float A2 = -expf(A_log[d * 4 + 2]);
    const float A3 = -expf(A_log[d * 4 + 3]);
    const float Dv = Dp[d];
    float h0 = 0.f, h1 = 0.f, h2 = 0.f, h3 = 0.f;
    const long long rbase = (long long)b * Ll;
    for (int l = 0; l < Ll; ++l) {
        const long long r = rbase + l;
        const float dl = delta[r * Dd + d];
        const float xv = xc[r * Dd + d];
        const float* bc = dbl + r * NP + dtRank;   // B at [0..3], C at [4..7]
        const float du = dl * xv;
        h0 = expf(dl * A0) * h0 + du * bc[0];
        h1 = expf(dl * A1) * h1 + du * bc[1];
        h2 = expf(dl * A2) * h2 + du * bc[2];
        h3 = expf(dl * A3) * h3 + du * bc[3];
        const float acc = h0 * bc[4] + h1 * bc[5] + h2 * bc[6] + h3 * bc[7];
        const float zv = xz[r * (2 * (long long)Dd) + Dd + d];
        y[r * Dd + d] = (acc + xv * Dv) * silu_f(zv);
    }
}

// ---- transpose (B, L, D) -> (B, D, L) -------------------------------------
__global__ void transpose_hl(const float* __restrict__ h, float* __restrict__ xt,
                             int Bb, int Ll, int Dd) {
    const long long idx = (long long)blockIdx.x * blockDim.x + threadIdx.x;
    const long long total = (long long)Bb * Ll * Dd;
    if (idx >= total) return;
    const int d = (int)(idx % Dd);
    const int l = (int)((idx / Dd) % Ll);
    const int b = (int)(idx / ((long long)Dd * Ll));
    xt[((long long)b * Dd + d) * Ll + l] = h[idx];
}

// ---- combine residuals + bn5: out = (x3 + 2h + x4^T) * g5[l] + b5[l] ------
__global__ void combine_kernel(const float* __restrict__ x3, const float* __restrict__ h,
                               const float* __restrict__ x4t, const float* __restrict__ g5,
                               const float* __restrict__ b5, float* __restrict__ out,
                               int Bb, int Ll, int Dd) {
    const long long idx = (long long)blockIdx.x * blockDim.x + threadIdx.x;
    const long long total = (long long)Bb * Ll * Dd;
    if (idx >= total) return;
    const int d = (int)(idx % Dd);
    const int l = (int)((idx / Dd) % Ll);
    const int b = (int)(idx / ((long long)Dd * Ll));
    const float v = x3[idx] + 2.0f * h[idx] + x4t[((long long)b * Dd + d) * Ll + l];
    out[idx] = v * g5[l] + b5[l];
}

// ---- final FC over flattened (L*16) ---------------------------------------
__global__ void fc_kernel(const float* __restrict__ h3, const float* __restrict__ W,
                          const float* __restrict__ bias, float* __restrict__ out,
                          int Bb, int KF, int NC) {
    const int idx = (int)(blockIdx.x * blockDim.x + threadIdx.x);
    if (idx >= Bb * NC) return;
    const int c = idx % NC;
    const int b = idx / NC;
    const float* hr = h3 + (size_t)b * KF;
    const float* wr = W + (size_t)c * KF;
    float acc = bias[c];
    for (int k = 0; k < KF; ++k) acc += hr[k] * wr[k];
    out[idx] = acc;
}

// ---------------------------------------------------------------------------
// Host orchestration
// ---------------------------------------------------------------------------
static inline unsigned cdiv(long long a, long long b) { return (unsigned)((a + b - 1) / b); }

static void launch_gemm(int NT, const float* A, const float* Wt, const float* bias,
                        float* C, int M, int N, int K, hipStream_t s) {
    const int tilesN = (NT == 2) ? N / 32 : N / 16;
    const unsigned blocks = (unsigned)((M / 256) * tilesN);
    if (NT == 2)
        gemm_wmma_tdm<2><<<blocks, 256, 0, s>>>(A, Wt, bias, C, M, N, K, tilesN);
    else
        gemm_wmma_tdm<1><<<blocks, 256, 0, s>>>(A, Wt, bias, C, M, N, K, tilesN);
}

struct MambaP {
    const float *ln_g, *ln_b, *in_proj, *conv_w, *conv_b, *x_proj, *dt_w, *dt_b, *A_log, *Dp, *out_proj;
};

static MambaP get_mamba(void* const* d_in, int base) {
    MambaP m;
    m.ln_g = (const float*)d_in[base + 0];  m.ln_b = (const float*)d_in[base + 1];
    m.in_proj = (const float*)d_in[base + 2];
    m.conv_w = (const float*)d_in[base + 3]; m.conv_b = (const float*)d_in[base + 4];
    m.x_proj = (const float*)d_in[base + 5];
    m.dt_w = (const float*)d_in[base + 6];  m.dt_b = (const float*)d_in[base + 7];
    m.A_log = (const float*)d_in[base + 8]; m.Dp = (const float*)d_in[base + 9];
    m.out_proj = (const float*)d_in[base + 10];
    return m;
}

static void run_mamba_block(const MambaP& mp, const float* xin, float* xout,
                            int Bb, int Ll, int Dd, int dtRank,
                            float* lnb, float* xzb, float* xcb, float* dblb, float* dltb,
                            hipStream_t s) {
    const long long rows = (long long)Bb * Ll;
    const int NP = dtRank + 8;  // dt_rank + 2*D_STATE
    layernorm_rows<<<cdiv(rows, 8), 256, 0, s>>>(xin, mp.ln_g, mp.ln_b, lnb, (int)rows, Dd);
    launch_gemm(2, lnb, mp.in_proj, nullptr, xzb, (int)rows, 2 * Dd, Dd, s);
    conv_silu<<<cdiv(rows * Dd, 256), 256, 0, s>>>(xzb, mp.conv_w, mp.conv_b, xcb, Bb, Ll, Dd);
    xproj_kernel<<<cdiv(rows * NP, 256), 256, 0, s>>>(xcb, mp.x_proj, dblb, rows, Dd, NP);
    delta_kernel<<<cdiv(rows * Dd, 256), 256, 0, s>>>(dblb, mp.dt_w, mp.dt_b, dltb, rows, NP, dtRank, Dd);
    mamba_scan<<<cdiv((long long)Bb * Dd, 256), 256, 0, s>>>(dltb, xcb, dblb, xzb, mp.A_log, mp.Dp,
                                                             lnb /* y reuses LN buffer */, Bb, Ll, Dd, NP, dtRank);
    launch_gemm((Dd % 32 == 0) ? 2 : 1, lnb, mp.out_proj, nullptr, xout, (int)rows, Dd, Dd, s);
}

extern "C" void kernel_launch(void* const* d_in, const int* in_sizes, int n_in,
                              void* d_out, int out_size, void* d_ws, size_t ws_size,
                              hipStream_t stream) {
    (void)in_sizes; (void)n_in; (void)out_size; (void)ws_size;
    const int B = 256, L = 176, CIN = 20, DM = 256, NC = 15;
    const long long R3 = (long long)B * L;    // 45056 rows, D=256
    const size_t S = (size_t)R3 * DM;         // == 65536*176 == 11,534,336 floats

    const float* x        = (const float*)d_in[0];
    const float* lin1_w   = (const float*)d_in[1];
    const float* lin1_b   = (const float*)d_in[2];
    const float* lin1_bng = (const float*)d_in[3];
    const float* lin1_bnb = (const float*)d_in[4];
    const float* lin5_bng = (const float*)d_in[93];
    const float* lin5_bnb = (const float*)d_in[94];
    const float* w1 = (const float*)d_in[95]; const float* b1 = (const float*)d_in[96];
    const float* w2 = (const float*)d_in[97]; const float* b2 = (const float*)d_in[98];
    const float* w3 = (const float*)d_in[99]; const float* b3 = (const float*)d_in[100];
    const float* fcw = (const float*)d_in[101]; const float* fcb = (const float*)d_in[102];

    float* W    = (float*)d_ws;
    float* bufH = W;            // h / res1                      [S]
    float* x3   = W + 1 * S;    // mamba3 chain (in-place)       [S]
    float* x4   = W + 2 * S;    // mamba4 chain (in-place)       [S]
    float* lnb  = W + 3 * S;    // LN out, reused as scan y      [S]
    float* xzb  = W + 4 * S;    // in_proj out (xi|z)            [2S]
    float* xcb  = W + 6 * S;    // conv+silu out                 [S]
    float* dltb = W + 7 * S;    // delta                         [S]
    float* dblb = W + 8 * S;    // x_proj out                    [R4*24 max]
    float* xcomb = xzb;         // head reuse (mamba done)
    float* h1 = xzb + S;        // R3*128
    float* h2 = xcb;            // R3*64
    float* h3 = dltb;           // R3*16

    // 1) lin1 + bn + leakyReLU -> bufH (res1)
    lin1_kernel<<<cdiv(R3 * DM, 256), 256, 0, stream>>>(x, lin1_w, lin1_b, lin1_bng, lin1_bnb,
                                                        bufH, B, L, CIN, DM);

    // 2) mamba3 chain: block0 reads bufH -> x3; blocks 1..3 in-place on x3
    for (int i = 0; i < 4; ++i) {
        MambaP mp = get_mamba(d_in, 5 + i * 11);
        run_mamba_block(mp, (i == 0) ? bufH : x3, x3, B, L, DM, 16,
                        lnb, xzb, xcb, dblb, dltb, stream);
    }

    // 3) mamba4 chain on transposed layout (B, 256, 176): seq=256, d_model=176
    transpose_hl<<<cdiv(R3 * DM, 256), 256, 0, stream>>>(bufH, x4, B, L, DM);
    for (int i = 0; i < 4; ++i) {
        MambaP mp = get_mamba(d_in, 49 + i * 11);
        run_mamba_block(mp, x4, x4, B, DM, L, 11,
                        lnb, xzb, xcb, dblb, dltb, stream);
    }

    // 4) combine: (x3 + res1) + (x4 + res2)^T = x3 + 2h + x4^T, then bn5
    combine_kernel<<<cdiv(R3 * DM, 256), 256, 0, stream>>>(x3, bufH, x4, lin5_bng, lin5_bnb,
                                                           xcomb, B, L, DM);

    // 5) head: 256 -> 128 -> 64 -> 16 (bias, no activation), then FC -> 15
    launch_gemm(2, xcomb, w1, b1, h1, (int)R3, 128, 256, stream);
    launch_gemm(2, h1, w2, b2, h2, (int)R3, 64, 128, stream);
    launch_gemm(1, h2, w3, b3, h3, (int)R3, 16, 64, stream);
    fc_kernel<<<cdiv((long long)B * NC, 256), 256, 0, stream>>>(h3, fcw, fcb, (float*)d_out,
                                                                B, L * 16, NC);
}